// BackWarp_9603546874299
// MI455X (gfx1250) — compile-verified
//
#include <hip/hip_runtime.h>

// dense_image_warp (TFA semantics): out[b,y,x,c] = bilinear(frame, y - flow_y, x - flow_x)
// floor clamped to [0, size-2], alphas clamped to [0,1].
//
// MI455X (gfx1250, wave32) bandwidth-bound gather kernel:
//  - 3D grid (x,y,b): no integer-division id decomposition
//  - 2 pixels/thread: flow read = one NT global_load_b128 (16B aligned),
//    output = 3x NT global_store_b64, corner gathers done per pixel
//  - frame gathers stay regular-temporal (4x spatial reuse; 1 image = 2.76 MB
//    lives in WGP$/L2; whole frame tensor (88.5 MB) < 192 MB L2)
//  - global_prefetch_b8 one image row ahead on the flow stream

#define BW_B 32
#define BW_H 360
#define BW_W 640
#define BW_C 3
#define TPB  160                 // 5 wave32 waves; 160 thr * 2 px = 320 px = half a row

typedef float v2f __attribute__((ext_vector_type(2)));
typedef float v4f __attribute__((ext_vector_type(4)));

__global__ __launch_bounds__(TPB) void backwarp_bilinear_kernel(
    const float* __restrict__ frame,   // [B,H,W,3]
    const float* __restrict__ flow,    // [B,H,W,2]
    float* __restrict__ out)           // [B,H,W,3]
{
    const int HW = BW_H * BW_W;

    const int p_row = blockIdx.x * TPB + threadIdx.x;  // pixel-pair index in row [0,320)
    const int y     = blockIdx.y;
    const int b     = blockIdx.z;
    const int x0    = p_row * 2;

    const int pix0 = b * HW + y * BW_W + x0;           // even
    const int pg   = pix0 >> 1;                        // pixel-pair global index

    // ---- streaming flow read: one NT b128 load (two pixels' dy,dx) ----
    const v4f* flow4 = (const v4f*)flow;
    const v4f f = __builtin_nontemporal_load(flow4 + pg);

    // prefetch the flow stream one image row ahead
    {
        const int PAIRS = (BW_B * HW) >> 1;
        const int pf = pg + (BW_W >> 1);
        if (pf < PAIRS) __builtin_prefetch((const void*)(flow4 + pf), 0, 0);
    }

    const size_t row_stride = (size_t)BW_W * BW_C;
    const float* __restrict__ img = frame + (size_t)b * HW * BW_C;

    float v[2][BW_C];

#pragma unroll
    for (int i = 0; i < 2; ++i) {
        // query point = grid - flow   (flow[...,0]=dy, flow[...,1]=dx)
        const float qy = (float)y        - f[2 * i + 0];
        const float qx = (float)(x0 + i) - f[2 * i + 1];

        // TFA clamp semantics
        float fy = floorf(qy);
        float fx = floorf(qx);
        fy = fminf(fmaxf(fy, 0.0f), (float)(BW_H - 2));
        fx = fminf(fmaxf(fx, 0.0f), (float)(BW_W - 2));
        const float ay = fminf(fmaxf(qy - fy, 0.0f), 1.0f);
        const float ax = fminf(fmaxf(qx - fx, 0.0f), 1.0f);
        const int iy = (int)fy;
        const int ix = (int)fx;

        // top-left corner texel; tl/tr are 6 contiguous floats,
        // bl/br are 6 contiguous floats one row below.
        const float* __restrict__ top = img + ((size_t)iy * BW_W + ix) * BW_C;
        const float* __restrict__ bot = top + row_stride;

#pragma unroll
        for (int c = 0; c < BW_C; ++c) {
            const float tl = top[c];
            const float tr = top[BW_C + c];
            const float bl = bot[c];
            const float br = bot[BW_C + c];
            const float it = fmaf(ax, tr - tl, tl);   // interp_top
            const float ib = fmaf(ax, br - bl, bl);   // interp_bot
            v[i][c] = fmaf(ay, ib - it, it);
        }
    }

    // write-once streaming output: 6 contiguous floats -> 3x NT b64 stores (8B aligned)
    v2f* o2 = (v2f*)(out + (size_t)pix0 * BW_C);
    v2f s0; s0.x = v[0][0]; s0.y = v[0][1];
    v2f s1; s1.x = v[0][2]; s1.y = v[1][0];
    v2f s2; s2.x = v[1][1]; s2.y = v[1][2];
    __builtin_nontemporal_store(s0, o2 + 0);
    __builtin_nontemporal_store(s1, o2 + 1);
    __builtin_nontemporal_store(s2, o2 + 2);
}

extern "C" void kernel_launch(void* const* d_in, const int* in_sizes, int n_in,
                              void* d_out, int out_size, void* d_ws, size_t ws_size,
                              hipStream_t stream) {
    (void)in_sizes; (void)n_in; (void)d_ws; (void)ws_size; (void)out_size;
    const float* frame = (const float*)d_in[0];  // [32,360,640,3] f32
    const float* flow  = (const float*)d_in[1];  // [32,360,640,2] f32
    float* out = (float*)d_out;                  // [32,360,640,3] f32

    // 640 px/row = 320 pairs = 2 blocks of 160 threads; grid covers (x, y, b) exactly.
    dim3 grid(BW_W / (2 * TPB), BW_H, BW_B);     // (2, 360, 32)
    dim3 block(TPB, 1, 1);
    backwarp_bilinear_kernel<<<grid, block, 0, stream>>>(frame, flow, out);
}